// Jpeg_23648089932316
// MI455X (gfx1250) — compile-verified
//
#include <hip/hip_runtime.h>

typedef __attribute__((ext_vector_type(2))) float v2f;
typedef __attribute__((ext_vector_type(8))) float v8f;

// JPEG quantization tables (row-major [u][v])
__constant__ float c_yq[64] = {
    16, 11, 10, 16, 24, 40, 51, 61,
    12, 12, 14, 19, 26, 58, 60, 55,
    14, 13, 16, 24, 40, 57, 69, 56,
    14, 17, 22, 29, 51, 87, 80, 62,
    18, 22, 37, 56, 68, 109, 103, 77,
    24, 35, 55, 64, 81, 104, 113, 92,
    49, 64, 78, 87, 103, 121, 120, 101,
    72, 92, 95, 98, 112, 100, 103, 99};
__constant__ float c_cq[64] = {
    17, 18, 24, 47, 99, 99, 99, 99,
    18, 21, 26, 66, 99, 99, 99, 99,
    24, 26, 56, 99, 99, 99, 99, 99,
    47, 66, 99, 99, 99, 99, 99, 99,
    99, 99, 99, 99, 99, 99, 99, 99,
    99, 99, 99, 99, 99, 99, 99, 99,
    99, 99, 99, 99, 99, 99, 99, 99,
    99, 99, 99, 99, 99, 99, 99, 99};

// One 16x16x16 f32 matmul as 4 chained K=4 WMMAs.
// A: constant matrix slices in A-layout (v2f per K-slice).
// b[2*i+j]: B-layout reg j of K-slice i (lane holds col = lane&15;
//           lanes0-15 -> K = 4i+j, lanes16-31 -> K = 4i+j+2).
__device__ __forceinline__ v8f wmma_k16(const v2f A[4], const float b[8], v8f c) {
#pragma unroll
  for (int i = 0; i < 4; ++i) {
    v2f B;
    B.x = b[2 * i];
    B.y = b[2 * i + 1];
    c = __builtin_amdgcn_wmma_f32_16x16x4_f32(false, A[i], false, B, (short)0, c,
                                              false, false);
  }
  return c;
}

// Full per-channel pipeline: pB is the (shifted) pixel tile in B-layout.
// Returns reconstructed (shift-domain) pixels in C-layout.
// Chain (left-mults only):  S1 = T*P ; S2 = T*S1^T = D^T ; S3 = Q'(S2) ;
//                           S4 = T^T*S3 ; S5 = T^T*S4^T = T^T*Q(D)*T.
__device__ __forceinline__ v8f dct_quant_idct(const float pB[8], const float tq[8],
                                              const v2f tA[4], const v2f tAT[4],
                                              float* __restrict__ lds, int lane) {
  const int h = lane >> 4;
  const int col = lane & 15;
  const v8f z = {0.f, 0.f, 0.f, 0.f, 0.f, 0.f, 0.f, 0.f};

  v8f s1 = wmma_k16(tA, pB, z);

  // ---- transpose #1 via LDS (store C-layout, read B-layout of S1^T) ----
#pragma unroll
  for (int r = 0; r < 8; ++r) lds[(r + 8 * h) * 20 + col] = s1[r];
  float bt[8];
#pragma unroll
  for (int rr = 0; rr < 8; ++rr) {
    const int i = rr >> 1, j = rr & 1;
    bt[rr] = lds[col * 20 + (4 * i + j + 2 * h)];
  }

  v8f d = wmma_k16(tA, bt, z);  // d = D^T (frequency domain, transposed)

  // ---- quantize / differentiable round / dequantize (transposed table) ----
#pragma unroll
  for (int r = 0; r < 8; ++r) {
    const float q = tq[r];
    const float x = d[r] * __builtin_amdgcn_rcpf(q);
    const float rn = __builtin_rintf(x);   // round-half-even == jnp.round
    const float t = x - rn;
    d[r] = (rn + t * t * t) * q;
  }

  // ---- cheap C-layout -> B-layout (half swaps via shfl_xor 16) ----
  float bb[8];
#pragma unroll
  for (int rr = 0; rr < 8; ++rr) {
    const int i = rr >> 1, j = rr & 1;
    if (i < 2) {
      const float lo = d[4 * i + j];
      const float sw = __shfl_xor(d[4 * i + j + 2], 16, 32);
      bb[rr] = (lane < 16) ? lo : sw;
    } else {
      const float hi = d[4 * (i - 2) + j + 2];
      const float sw = __shfl_xor(d[4 * (i - 2) + j], 16, 32);
      bb[rr] = (lane < 16) ? sw : hi;
    }
  }

  v8f e = wmma_k16(tAT, bb, z);

  // ---- transpose #2 via LDS ----
#pragma unroll
  for (int r = 0; r < 8; ++r) lds[(r + 8 * h) * 20 + col] = e[r];
  float bt2[8];
#pragma unroll
  for (int rr = 0; rr < 8; ++rr) {
    const int i = rr >> 1, j = rr & 1;
    bt2[rr] = lds[col * 20 + (4 * i + j + 2 * h)];
  }

  return wmma_k16(tAT, bt2, z);
}

__global__ __launch_bounds__(256) void jpeg_wmma_kernel(
    const float* __restrict__ in, float* __restrict__ out, int n_tiles) {
  __shared__ float lds_all[8 * 320];  // 16 rows * stride 20 per wave (conflict-free)
  const int lane = threadIdx.x & 31;
  const int wid = threadIdx.x >> 5;
  float* lds = &lds_all[wid * 320];

  const int tile = blockIdx.x * 8 + wid;
  if (tile >= n_tiles) return;
  const int b = tile >> 10;          // 1024 tiles per image (32x32 tiles of 16x16)
  const int tp = tile & 1023;
  const int row0 = (tp >> 5) << 4;
  const int col0 = (tp & 31) << 4;
  const int m = lane & 15, h = lane >> 4, col = lane & 15;

  // ---- DCT constants in A-layout: T = diag(A8,A8), A8[u][x]=alpha(u)/2*cos((2x+1)u*pi/16)
  v2f tA[4], tAT[4];
#pragma unroll
  for (int i = 0; i < 4; ++i) {
#pragma unroll
    for (int j = 0; j < 2; ++j) {
      const int k = 4 * i + j + 2 * h;
      const bool same = ((m ^ k) & 8) == 0;  // block-diagonal
      const int mu = m & 7, kx = k & 7;
      const float aF = (mu == 0 ? 0.70710678f : 1.0f) * 0.5f *
                       __cosf((float)((2 * kx + 1) * mu) * 0.19634954f);  // T[m][k]
      const float aB = (kx == 0 ? 0.70710678f : 1.0f) * 0.5f *
                       __cosf((float)((2 * mu + 1) * kx) * 0.19634954f);  // T^T[m][k]
      const float vF = same ? aF : 0.0f;
      const float vB = same ? aB : 0.0f;
      if (j == 0) { tA[i].x = vF; tAT[i].x = vB; }
      else        { tA[i].y = vF; tAT[i].y = vB; }
    }
  }

  // Per-lane quant tables, pre-transposed: element (reg r, lane) of D^T needs qt[col&7][r]
  float tqy[8], tqc[8];
#pragma unroll
  for (int r = 0; r < 8; ++r) {
    tqy[r] = c_yq[(lane & 7) * 8 + r];
    tqc[r] = c_cq[(lane & 7) * 8 + r];
  }

  const long plane = 512L * 512L;
  const float* pr = in + (long)b * 3 * plane;
  const float* pg = pr + plane;
  const float* pbl = pg + plane;

  // ---- load RGB tile in B-layout, fused [-1,1]->[0,255] + RGB->YCbCr (+shift/center)
  float yB[8], cbB[8], crB[8];
#pragma unroll
  for (int rr = 0; rr < 8; ++rr) {
    const int k = 4 * (rr >> 1) + (rr & 1) + 2 * h;
    const long off = (long)(row0 + k) * 512 + col0 + col;
    const float r = pr[off], g = pg[off], bl = pbl[off];
    // x*127.5+127.5 folded into linear transforms; Y shift(0)-128, C shift(128)-128=0
    yB[rr] = 127.5f * (0.299f * r + 0.587f * g + 0.114f * bl) - 0.5f;
    cbB[rr] = 127.5f * (-0.168736f * r - 0.331264f * g + 0.5f * bl);
    crB[rr] = 127.5f * (0.5f * r - 0.418688f * g - 0.081312f * bl);
  }

  const v8f Y = dct_quant_idct(yB, tqy, tA, tAT, lds, lane);
  const v8f Cb = dct_quant_idct(cbB, tqc, tA, tAT, lds, lane);
  const v8f Cr = dct_quant_idct(crB, tqc, tA, tAT, lds, lane);

  // ---- YCbCr -> RGB, clip, /255, store from C-layout
  float* qr = out + (long)b * 3 * plane;
  float* qg = qr + plane;
  float* qb = qg + plane;
#pragma unroll
  for (int r = 0; r < 8; ++r) {
    const float yy = Y[r] + 128.0f;
    const float cb = Cb[r], cr = Cr[r];
    float R = yy + 1.402f * cr;
    float G = yy - 0.344136f * cb - 0.714136f * cr;
    float B = yy + 1.772f * cb;
    R = fminf(fmaxf(R, 0.f), 255.f) * (1.0f / 255.0f);
    G = fminf(fmaxf(G, 0.f), 255.f) * (1.0f / 255.0f);
    B = fminf(fmaxf(B, 0.f), 255.f) * (1.0f / 255.0f);
    const long off = (long)(row0 + r + 8 * h) * 512 + col0 + col;
    qr[off] = R;
    qg[off] = G;
    qb[off] = B;
  }
}

extern "C" void kernel_launch(void* const* d_in, const int* in_sizes, int n_in,
                              void* d_out, int out_size, void* d_ws, size_t ws_size,
                              hipStream_t stream) {
  (void)n_in; (void)d_ws; (void)ws_size; (void)out_size;
  const float* img = (const float*)d_in[0];
  float* out = (float*)d_out;
  const int n = in_sizes[0];             // b*3*512*512
  const int n_tiles = n / (3 * 16 * 16); // one wave per 16x16x3 tile
  const int blocks = (n_tiles + 7) / 8;  // 8 waves (256 threads) per block
  jpeg_wmma_kernel<<<blocks, 256, 0, stream>>>(img, out, n_tiles);
}